// PennyLaneSimpleClassifier_3788161155353
// MI455X (gfx1250) — compile-verified
//
#include <hip/hip_runtime.h>

#define NQ      12
#define DIM     4096        // 2^12
#define NPAIR   2048
#define TPB     256
#define NLAYERS 4
#define NOUT    10
#define PI_F    3.14159265358979323846f

typedef float v2f __attribute__((ext_vector_type(2)));
typedef float v8f __attribute__((ext_vector_type(8)));

// Apply the Gray-code map l times: physical index of logical j after l CNOT chains.
__device__ __forceinline__ int grayl(int v, int l)
{
    for (int i = 0; i < l; ++i) v ^= (v >> 1);
    return v;
}

// Apply a general complex 2x2 gate U on logical bit position bp, given that the
// state is stored physically permuted by g^l (l prior CNOT chains).
// Each pair is fully owned by one thread; barrier at end.
__device__ __forceinline__ void gate2x2(float2* st, int tid, int s, int l,
                                        float u00r, float u00i, float u01r, float u01i,
                                        float u10r, float u10i, float u11r, float u11i)
{
    const int m = grayl(s, l);            // physical XOR mask (g^l is linear)
#pragma unroll
    for (int it = 0; it < NPAIR / TPB; ++it) {
        int p  = tid + it * TPB;
        int j0 = ((p & ~(s - 1)) << 1) | (p & (s - 1));  // logical index, bit bp = 0
        int v0 = grayl(j0, l);                            // physical location
        int v1 = v0 ^ m;
        float2 a0 = st[v0];
        float2 a1 = st[v1];
        float2 n0, n1;
        n0.x = u00r * a0.x - u00i * a0.y + u01r * a1.x - u01i * a1.y;
        n0.y = u00r * a0.y + u00i * a0.x + u01r * a1.y + u01i * a1.x;
        n1.x = u10r * a0.x - u10i * a0.y + u11r * a1.x - u11i * a1.y;
        n1.y = u10r * a0.y + u10i * a0.x + u11r * a1.y + u11i * a1.x;
        st[v0] = n0;
        st[v1] = n1;
    }
    __syncthreads();
}

// One workgroup = one batch element. State lives in LDS (32 KB, float2 packed).
__global__ void __launch_bounds__(TPB)
qsim_kernel(const float* __restrict__ x, const float* __restrict__ theta,
            float* __restrict__ qout)
{
    __shared__ float2 st[DIM];
    __shared__ float  zacc[NQ];

    const int b   = blockIdx.x;
    const int tid = threadIdx.x;

#pragma unroll
    for (int it = 0; it < DIM / TPB; ++it)
        st[tid + it * TPB] = make_float2(0.f, 0.f);
    if (tid < NQ) zacc[tid] = 0.f;
    if (tid == 0) st[0] = make_float2(1.f, 0.f);
    __syncthreads();

    // ---- RY angle embedding: RY(x*pi) on wire q (wire 0 = MSB); no permutation yet ----
    for (int q = 0; q < NQ; ++q) {
        float h = x[b * NQ + q] * PI_F * 0.5f;
        float s, c;
        __sincosf(h, &s, &c);
        gate2x2(st, tid, 1 << (NQ - 1 - q), 0,
                c, 0.f, -s, 0.f,
                s, 0.f,  c, 0.f);
    }

    // ---- 4 layers of fused RZ(a)->RX(bt)->RZ(c); the CNOT chain after each layer
    //      is the Gray-code permutation, folded into addressing (zero data motion) ----
    int p = 0;
    for (int l = 0; l < NLAYERS; ++l) {
        for (int q = 0; q < NQ; ++q) {
            float a2 = theta[p] * 0.5f;
            float b2 = theta[p + 1] * 0.5f;
            float c2 = theta[p + 2] * 0.5f;
            p += 3;
            float sb, cb, ss, cs, sd, cd;
            __sincosf(b2, &sb, &cb);
            __sincosf(a2 + c2, &ss, &cs);
            __sincosf(a2 - c2, &sd, &cd);
            // U = Rz(c)*Rx(bt)*Rz(a):
            //   U00 = cb*e^{-i(a2+c2)}        U01 = sb*( sin d - i cos d), d = a2-c2
            //   U10 = sb*(-sin d - i cos d)   U11 = cb*e^{+i(a2+c2)}
            gate2x2(st, tid, 1 << (NQ - 1 - q), l,
                     cb * cs, -cb * ss,
                     sb * sd, -sb * cd,
                    -sb * sd, -sb * cd,
                     cb * cs,  cb * ss);
        }
        // CNOT chain: handled implicitly by increasing l (physical = g^l(logical))
    }

    // ---- <Z_i>: iterate logical index j, read physical g^4(j) ----
    float z[NQ];
#pragma unroll
    for (int i = 0; i < NQ; ++i) z[i] = 0.f;
#pragma unroll
    for (int it = 0; it < DIM / TPB; ++it) {
        int    j = tid + it * TPB;
        int    v = grayl(j, NLAYERS);
        float2 a = st[v];
        float pr = a.x * a.x + a.y * a.y;
#pragma unroll
        for (int i = 0; i < NQ; ++i)
            z[i] += ((j >> (NQ - 1 - i)) & 1) ? -pr : pr;
    }
#pragma unroll
    for (int i = 0; i < NQ; ++i) atomicAdd(&zacc[i], z[i]);
    __syncthreads();
    if (tid < NQ) qout[b * NQ + tid] = zacc[tid];
}

// out[B,10] = qout[B,12] @ W^T + b  using V_WMMA_F32_16X16X4_F32.
// One wave per 16x16 output tile; K=12 as three K=4 accumulating WMMAs.
__global__ void __launch_bounds__(256)
linear_wmma_kernel(const float* __restrict__ qout, const float* __restrict__ W,
                   const float* __restrict__ bias, float* __restrict__ out, int tiles)
{
    int gwave = (blockIdx.x * blockDim.x + threadIdx.x) >> 5; // tile id (uniform per wave)
    int lane  = threadIdx.x & 31;
    if (gwave >= tiles) return;                               // whole-wave uniform exit

    int   n     = lane & 15;                 // output column (N) this lane owns in B/C/D
    int   hi    = lane >> 4;                 // 0: K0/K1 group, 1: K2/K3 group (A & B layouts)
    int   koff  = hi * 2;
    int   arow  = gwave * 16 + n;            // A-matrix row for this lane (M = lane%16)
    int   nc    = (n < NOUT) ? n : (NOUT - 1);
    float valid = (n < NOUT) ? 1.f : 0.f;    // branch-free column masking

    v8f acc = {};
#pragma unroll
    for (int kk = 0; kk < 3; ++kk) {
        int kbase = kk * 4 + koff;
        v2f a, bm;
        a.x  = qout[arow * NQ + kbase];
        a.y  = qout[arow * NQ + kbase + 1];
        bm.x = W[nc * NQ + kbase]     * valid;   // B[k][n] = W[n][k], zeroed for n>=10
        bm.y = W[nc * NQ + kbase + 1] * valid;
        // (neg_a, A, neg_b, B, c_mod, C, reuse_a, reuse_b)
        acc = __builtin_amdgcn_wmma_f32_16x16x4_f32(false, a, false, bm,
                                                    (short)0, acc, false, false);
    }

    float bv = bias[nc];
#pragma unroll
    for (int v = 0; v < 8; ++v) {
        int row = gwave * 16 + v + hi * 8;   // C/D layout: VGPR v -> M=v (lo) / v+8 (hi)
        if (n < NOUT) out[row * NOUT + n] = acc[v] + bv;
    }
}

extern "C" void kernel_launch(void* const* d_in, const int* in_sizes, int n_in,
                              void* d_out, int out_size, void* d_ws, size_t ws_size,
                              hipStream_t stream)
{
    const float* x     = (const float*)d_in[0];  // [B, 12]
    const float* theta = (const float*)d_in[1];  // [144]
    const float* W     = (const float*)d_in[2];  // [10, 12]
    const float* bias  = (const float*)d_in[3];  // [10]
    float*       out   = (float*)d_out;          // [B, 10]
    float*       qout  = (float*)d_ws;           // [B, 12] scratch

    int B = in_sizes[0] / NQ;                    // 4096

    qsim_kernel<<<B, TPB, 0, stream>>>(x, theta, qout);

    int tiles  = B / 16;                         // 256
    int blocks = (tiles + 7) / 8;                // 8 waves per block
    linear_wmma_kernel<<<blocks, 256, 0, stream>>>(qout, W, bias, out, tiles);
}